// NMS_3590592659705
// MI455X (gfx1250) — compile-verified
//
#include <hip/hip_runtime.h>
#include <stdint.h>

typedef unsigned int uint32;
typedef uint32 u32x4 __attribute__((ext_vector_type(4)));
typedef int    i32x4 __attribute__((ext_vector_type(4)));
typedef int    i32x8 __attribute__((ext_vector_type(8)));

#define IMG_W    1024
#define IMG_H    1024
#define N_IMG    32
#define ROWS_OUT 4
#define TILE_ROWS 10   // ROWS_OUT + 6 halo
#define NBINS    2048
#define CAP      2048
#define TOPK     200
#define THRESH   0.1f

// ---------------------------------------------------------------------------
// Zero scratch (histograms + counters) -- required each launch (determinism)
// ---------------------------------------------------------------------------
__global__ void nms_zero(uint32* __restrict__ p, int n) {
  int i = blockIdx.x * blockDim.x + threadIdx.x;
  if (i < n) p[i] = 0u;
}

// ---------------------------------------------------------------------------
// Peak detection. COLLECT==0: build per-image value histogram of peaks.
// COLLECT==1: append keys for peaks with bin >= per-image cutoff.
// Tile = 4 output rows x 1024 cols; 10 input rows DMA'd into LDS via the
// CDNA5 Tensor Data Mover (tensor_load_to_lds + s_wait_tensorcnt).
// Zero padding is equivalent to -inf padding since all data is in [0,1).
// ---------------------------------------------------------------------------
template <int COLLECT>
__global__ __launch_bounds__(256)
void nms_detect(const float* __restrict__ src,
                uint32* __restrict__ hist,
                uint32* __restrict__ counter,
                const uint32* __restrict__ cutoffBin,
                unsigned long long* __restrict__ keys)
{
  __shared__ float ldsIn[TILE_ROWS * IMG_W];   // 40 KB raw tile (+halo)
  __shared__ float ldsV[ROWS_OUT * IMG_W];     // 16 KB vertical 7-max
  __shared__ uint32 sCut;

  const int tid = threadIdx.x;
  const int img = blockIdx.x >> 8;          // 256 row-blocks per image
  const int rb  = blockIdx.x & 255;
  const int r0  = rb * ROWS_OUT;

  const int padTop   = (r0 < 3) ? (3 - r0) : 0;
  const int lastRow  = r0 + ROWS_OUT - 1 + 3;
  const int padBot   = (lastRow > IMG_H - 1) ? (lastRow - (IMG_H - 1)) : 0;
  const int loadRow0 = r0 - 3 + padTop;
  const int loadRows = TILE_ROWS - padTop - padBot;

  // Zero-fill halo rows the DMA won't write (image top/bottom edges only).
  for (int i = tid; i < padTop * IMG_W; i += 256) ldsIn[i] = 0.0f;
  {
    const int base = (TILE_ROWS - padBot) * IMG_W;
    for (int i = tid; i < padBot * IMG_W; i += 256) ldsIn[base + i] = 0.0f;
  }

  if (tid == 0) {
    if (COLLECT) sCut = cutoffBin[img];
    // ---- Tensor DMA descriptor (D#), per CDNA5 ISA ch.8 bit layout ----
    unsigned long long gaddr = (unsigned long long)(const void*)src
        + ((unsigned long long)img * (unsigned long long)(IMG_W * IMG_H)
           + (unsigned long long)loadRow0 * IMG_W) * 4ull;
    uint32 ldsByte = (uint32)(unsigned long long)(void*)ldsIn
                   + (uint32)(padTop * IMG_W * 4);
    u32x4 g0;
    g0[0] = 1u;                                            // count=1, user mode
    g0[1] = ldsByte;                                       // lds_addr (bytes)
    g0[2] = (uint32)gaddr;                                 // global_addr[31:0]
    g0[3] = ((uint32)(gaddr >> 32) & 0x01FFFFFFu)          // global_addr[56:32]
            | (2u << 30);                                  // type = 2 ("image")
    i32x8 g1;
    g1[0] = (int)(2u << 16);                               // data_size=2 (4B), no mask/pad
    g1[1] = (int)((uint32)IMG_W << 16);                    // tensor_dim0 lo16 (=1024)
    g1[2] = 0;                                             // tensor_dim0 hi | tensor_dim1 lo
    g1[3] = (int)(0x10u | ((uint32)IMG_W << 16));          // tensor_dim1 hi (big) | tile_dim0=1024
    g1[4] = loadRows;                                      // tile_dim1 | tile_dim2=0
    g1[5] = IMG_W;                                         // tensor_dim0_stride lo32
    g1[6] = 0;
    g1[7] = 0;
    i32x4 gz4 = {0, 0, 0, 0};                              // groups 2/3 unused (2D)
    i32x8 gz8 = {0, 0, 0, 0, 0, 0, 0, 0};                  // extra group (clang-23 form)
    __builtin_amdgcn_tensor_load_to_lds(g0, g1, gz4, gz4, gz8, 0);
    __builtin_amdgcn_s_wait_tensorcnt(0);
  }
  __syncthreads();

  // Stage B: vertical 7-max -> ldsV (separable pool, part 1)
  for (int p = tid; p < ROWS_OUT * IMG_W; p += 256) {
    const int rout = p >> 10;
    const int c = p & (IMG_W - 1);
    float m = ldsIn[rout * IMG_W + c];
    #pragma unroll
    for (int dr = 1; dr < 7; ++dr) m = fmaxf(m, ldsIn[(rout + dr) * IMG_W + c]);
    ldsV[p] = m;
  }
  __syncthreads();

  // Stage C: horizontal 7-max, compare with center, histogram/collect
  for (int p = tid; p < ROWS_OUT * IMG_W; p += 256) {
    const int rout = p >> 10;
    const int c = p & (IMG_W - 1);
    const float center = ldsIn[(rout + 3) * IMG_W + c];
    float pooled = 0.0f;
    #pragma unroll
    for (int dc = -3; dc <= 3; ++dc) {
      const int cc = c + dc;
      if (cc >= 0 && cc < IMG_W) pooled = fmaxf(pooled, ldsV[(rout << 10) + cc]);
    }
    if (pooled == center && center > THRESH) {
      int bin = (int)((center - THRESH) * ((float)NBINS / 0.9f));
      bin = bin < 0 ? 0 : (bin > NBINS - 1 ? NBINS - 1 : bin);
      if (COLLECT) {
        if (bin >= (int)sCut) {
          uint32 pos = atomicAdd(&counter[img], 1u);
          if (pos < CAP) {
            uint32 idxg = ((uint32)(r0 + rout) << 10) | (uint32)c;
            // value desc, then index asc == key desc
            keys[((unsigned long long)img << 11) + pos] =
                ((unsigned long long)__float_as_uint(center) << 32)
              | (unsigned long long)(~idxg);
          }
        }
      } else {
        atomicAdd(&hist[(img << 11) + bin], 1u);
      }
    }
  }
}

// ---------------------------------------------------------------------------
// Per-image cutoff: largest bin b with suffix-count(b) >= TOPK
// ---------------------------------------------------------------------------
__global__ __launch_bounds__(256)
void nms_cutoff(const uint32* __restrict__ hist, uint32* __restrict__ cutoffBin)
{
  __shared__ uint32 h[NBINS];
  __shared__ uint32 csum[256];
  const int tid = threadIdx.x;
  const int img = blockIdx.x;
  for (int i = tid; i < NBINS; i += 256) h[i] = hist[(img << 11) + i];
  __syncthreads();
  uint32 s = 0;
  #pragma unroll
  for (int j = 0; j < 8; ++j) s += h[tid * 8 + j];
  csum[tid] = s;
  __syncthreads();
  if (tid == 0) {
    uint32 acc = 0, cut = 0;
    for (int t = 255; t >= 0; --t) {
      if (acc + csum[t] >= TOPK) {
        for (int b = t * 8 + 7; b >= t * 8; --b) {
          acc += h[b];
          if (acc >= TOPK) { cut = (uint32)b; break; }
        }
        break;
      }
      acc += csum[t];
    }
    cutoffBin[img] = cut;  // 0 if <TOPK peaks total (can't happen here)
  }
}

// ---------------------------------------------------------------------------
// Per-image bitonic sort (descending on packed keys), emit coords + probs
// ---------------------------------------------------------------------------
__global__ __launch_bounds__(1024)
void nms_sort(const uint32* __restrict__ counter,
              const unsigned long long* __restrict__ keys,
              float* __restrict__ out)
{
  __shared__ unsigned long long k[CAP];    // 16 KB
  const int tid = threadIdx.x;
  const int img = blockIdx.x;
  uint32 cnt = counter[img];
  if (cnt > CAP) cnt = CAP;
  for (int i = tid; i < CAP; i += 1024)
    k[i] = (i < (int)cnt) ? keys[((unsigned long long)img << 11) + i] : 0ull;
  __syncthreads();
  for (int ksz = 2; ksz <= CAP; ksz <<= 1) {
    for (int j = ksz >> 1; j > 0; j >>= 1) {
      for (int i = tid; i < CAP; i += 1024) {
        const int p = i ^ j;
        if (p > i) {
          const bool up = ((i & ksz) == 0);
          unsigned long long a = k[i], b = k[p];
          if (up ? (a < b) : (a > b)) { k[i] = b; k[p] = a; }
        }
      }
      __syncthreads();
    }
  }
  if (tid < TOPK) {
    unsigned long long key = k[tid];
    const float  v    = __uint_as_float((uint32)(key >> 32));
    const uint32 idxg = ~((uint32)key);
    out[(img * TOPK + tid) * 2 + 0] = (float)(idxg >> 10);         // row
    out[(img * TOPK + tid) * 2 + 1] = (float)(idxg & (IMG_W - 1)); // col
    out[N_IMG * TOPK * 2 + img * TOPK + tid] = v;                  // prob
  }
}

// ---------------------------------------------------------------------------
extern "C" void kernel_launch(void* const* d_in, const int* in_sizes, int n_in,
                              void* d_out, int out_size, void* d_ws, size_t ws_size,
                              hipStream_t stream)
{
  (void)in_sizes; (void)n_in; (void)out_size; (void)ws_size;
  const float* src = (const float*)d_in[0];
  uint8_t* ws = (uint8_t*)d_ws;
  // Workspace layout (~0.77 MB total):
  uint32* hist    = (uint32*)ws;                         // 32*2048*4 = 262144 B
  uint32* counter = (uint32*)(ws + 262144);              // 32*4      (pad 128)
  uint32* cutoff  = (uint32*)(ws + 262144 + 128);        // 32*4      (pad 128)
  unsigned long long* keys =
      (unsigned long long*)(ws + 262144 + 256);          // 32*2048*8 = 524288 B

  const int nzero = 65600;                               // hist + counter + cutoff
  nms_zero<<<(nzero + 255) / 256, 256, 0, stream>>>((uint32*)ws, nzero);
  nms_detect<0><<<N_IMG * 256, 256, 0, stream>>>(src, hist, counter, cutoff, keys);
  nms_cutoff<<<N_IMG, 256, 0, stream>>>(hist, cutoff);
  nms_detect<1><<<N_IMG * 256, 256, 0, stream>>>(src, hist, counter, cutoff, keys);
  nms_sort<<<N_IMG, 1024, 0, stream>>>(counter, keys, (float*)d_out);
}